// AdaptiveInput_27745488732194
// MI455X (gfx1250) — compile-verified
//
#include <hip/hip_runtime.h>
#include <hip/hip_bf16.h>

typedef __attribute__((ext_vector_type(2))) float v2f;
typedef __attribute__((ext_vector_type(8))) float v8f;

#define D_MODEL 1024

// ---------------------------------------------------------------------------
// Phase 1: bucketize tokens, compact (token_pos, local_id) per bucket.
// ---------------------------------------------------------------------------
__global__ void bucketize_kernel(const int* __restrict__ ids, int n,
                                 int* __restrict__ counts,
                                 int* __restrict__ tok,   // [4][n]
                                 int* __restrict__ loc)   // [4][n]
{
    int t = blockIdx.x * blockDim.x + threadIdx.x;
    if (t >= n) return;
    int id = ids[t];
    int b, lo;
    if (id < 20000)       { b = 0; lo = 0; }
    else if (id < 40000)  { b = 1; lo = 20000; }
    else if (id < 200000) { b = 2; lo = 40000; }
    else                  { b = 3; lo = 200000; }
    int p = atomicAdd(&counts[b], 1);
    tok[(size_t)b * n + p] = t;
    loc[(size_t)b * n + p] = id - lo;
}

// ---------------------------------------------------------------------------
// Phase 2a: bucket 0 is a plain 1024-wide gather (4KB per row, float4).
// ---------------------------------------------------------------------------
__global__ __launch_bounds__(256) void copy0_kernel(
    const float* __restrict__ E0, const int* __restrict__ counts,
    const int* __restrict__ tok, const int* __restrict__ loc,
    float* __restrict__ out)
{
    int i = blockIdx.x;
    if (i >= counts[0]) return;
    int t = tok[i];
    int l = loc[i];
    const float4* src = (const float4*)(E0 + (size_t)l * D_MODEL);
    float4*       dst = (float4*)(out + (size_t)t * D_MODEL);
    dst[threadIdx.x] = src[threadIdx.x];   // 256 threads x 16B = 4KB row
}

// ---------------------------------------------------------------------------
// Phase 2b: grouped GEMM per bucket using fp32 WMMA 16x16x4.
//   A = gathered embeddings [count x K]  (staged in LDS, padded stride K+4)
//   B = W^T  (W is [1024 x K] row-major -> B[k,n] = W[n,k])
//   D = [count x 1024] scattered to out rows by token position.
// Block: 256 threads = 8 waves. Block covers RT row-tiles (RT*16 tokens) and
// all 64 column tiles (8 waves x 8 tiles). Each B fragment is reused RT times.
// ---------------------------------------------------------------------------
template <int K, int RT>
__global__ __launch_bounds__(256) void gemm_bucket_kernel(
    const float* __restrict__ E, const float* __restrict__ W,
    const int* __restrict__ counts, int bucket,
    const int* __restrict__ tok, const int* __restrict__ loc,
    float* __restrict__ out)
{
    constexpr int ROWS   = RT * 16;
    constexpr int LDSTR  = K + 4;            // padded row stride (floats)
    __shared__ __align__(16) float Alds[ROWS * LDSTR];
    __shared__ int stok[ROWS];
    __shared__ int sloc[ROWS];

    const int count = counts[bucket];
    const int base  = blockIdx.x * ROWS;
    if (base >= count) return;               // uniform across the block

    const int tid = threadIdx.x;
    if (tid < ROWS) {
        int idx = base + tid;
        stok[tid] = (idx < count) ? tok[idx] : 0;
        sloc[tid] = (idx < count) ? loc[idx] : 0;  // row 0: safe dummy gather
    }
    __syncthreads();

    // Stage A tile: ROWS gathered rows of K floats, float4-vectorized.
    constexpr int KV = K / 4;                // float4s per row
    for (int i = tid; i < ROWS * KV; i += 256) {
        int r = i / KV, c = i % KV;
        float4 v = ((const float4*)(E + (size_t)sloc[r] * K))[c];
        ((float4*)Alds)[r * (LDSTR / 4) + c] = v;
    }
    __syncthreads();

    const int lane = tid & 31;
    const int wave = tid >> 5;
    const int m    = lane & 15;              // M (A) / N (B,D) within tile
    const int khi  = lane >> 4;              // selects K pair within step

    // Hoist per-lane output row offsets (do NOT depend on the n-tile loop):
    // lane stores D rows mm = j + 8*khi of each row-tile.
    int rowoff[RT * 8];
#pragma unroll
    for (int rt = 0; rt < RT; ++rt)
#pragma unroll
        for (int j = 0; j < 8; ++j)
            rowoff[rt * 8 + j] = stok[rt * 16 + j + 8 * khi] * D_MODEL;

    const bool full = (base + ROWS <= count);   // block-uniform

    for (int ntl = 0; ntl < 8; ++ntl) {
        const int n0 = (wave * 8 + ntl) * 16;

        v8f acc[RT];
#pragma unroll
        for (int rt = 0; rt < RT; ++rt) acc[rt] = (v8f){0,0,0,0,0,0,0,0};

        // lane's B column base: B[k, n0+m] = W[(n0+m)*K + k]
        const float* wp = W + (size_t)(n0 + m) * K + 2 * khi;

#pragma unroll 4
        for (int s = 0; s < K / 4; ++s) {
            v2f b = *(const v2f*)(wp + 4 * s);      // global_load_b64
#pragma unroll
            for (int rt = 0; rt < RT; ++rt) {
                const float* ap = &Alds[(rt * 16 + m) * LDSTR + 4 * s + 2 * khi];
                v2f a = *(const v2f*)ap;            // ds_load_b64, bank-clean
                acc[rt] = __builtin_amdgcn_wmma_f32_16x16x4_f32(
                    /*neg_a=*/false, a, /*neg_b=*/false, b,
                    /*c_mod=*/(short)0, acc[rt],
                    /*reuse_a=*/false, /*reuse_b=*/false);
            }
        }

        const int ncol = n0 + m;
        if (full) {
            // Straight-line stores, no divergence, no exec-mask ladder.
#pragma unroll
            for (int rt = 0; rt < RT; ++rt)
#pragma unroll
                for (int j = 0; j < 8; ++j)
                    out[(size_t)(rowoff[rt * 8 + j] + ncol)] = acc[rt][j];
        } else {
            // Boundary block only: guard invalid rows (dummy token 0).
#pragma unroll
            for (int rt = 0; rt < RT; ++rt) {
                const int rb = base + rt * 16;
#pragma unroll
                for (int j = 0; j < 8; ++j)
                    if (rb + j + 8 * khi < count)
                        out[(size_t)(rowoff[rt * 8 + j] + ncol)] = acc[rt][j];
            }
        }
    }
}

// ---------------------------------------------------------------------------
extern "C" void kernel_launch(void* const* d_in, const int* in_sizes, int n_in,
                              void* d_out, int out_size, void* d_ws, size_t ws_size,
                              hipStream_t stream)
{
    const int*   ids = (const int*)d_in[0];
    const float* E0  = (const float*)d_in[1];
    const float* E1  = (const float*)d_in[2];
    const float* E2  = (const float*)d_in[3];
    const float* E3  = (const float*)d_in[4];
    const float* W1  = (const float*)d_in[5];
    const float* W2  = (const float*)d_in[6];
    const float* W3  = (const float*)d_in[7];
    float* out = (float*)d_out;

    const int n = in_sizes[0];               // 32768 tokens

    char* ws     = (char*)d_ws;
    int*  counts = (int*)ws;                                  // 4 ints
    int*  tok    = (int*)(ws + 256);                          // [4][n]
    int*  loc    = (int*)(ws + 256 + 4 * (size_t)n * sizeof(int));

    hipMemsetAsync(counts, 0, 4 * sizeof(int), stream);

    bucketize_kernel<<<(n + 255) / 256, 256, 0, stream>>>(ids, n, counts, tok, loc);

    copy0_kernel<<<n, 256, 0, stream>>>(E0, counts, tok, loc, out);

    // bucket 1: K=512, 16 rows/block  -> LDS ~33KB
    gemm_bucket_kernel<512, 1><<<(n + 15) / 16, 256, 0, stream>>>(
        E1, W1, counts, 1, tok + (size_t)n, loc + (size_t)n, out);
    // bucket 2: K=256, 32 rows/block
    gemm_bucket_kernel<256, 2><<<(n + 31) / 32, 256, 0, stream>>>(
        E2, W2, counts, 2, tok + 2 * (size_t)n, loc + 2 * (size_t)n, out);
    // bucket 3: K=128, 64 rows/block
    gemm_bucket_kernel<128, 4><<<(n + 63) / 64, 256, 0, stream>>>(
        E3, W3, counts, 3, tok + 3 * (size_t)n, loc + 3 * (size_t)n, out);
}